// DiffMarchingCubes_9912784519488
// MI455X (gfx1250) — compile-verified
//
#include <hip/hip_runtime.h>
#include <hip/hip_bf16.h>
#include <cstdint>

// Wait until at most `n` async (global<->LDS) ops remain outstanding for this
// wave. Inline asm with a "memory" clobber: acts as a compiler ordering
// barrier so the subsequent LDS read of the staged buffer cannot be hoisted
// above the wait (a builtin modeled as inaccessible-mem side effects would not
// guarantee that).
#define WAIT_ASYNC(n) asm volatile("s_wait_asynccnt " #n ::: "memory")

// gfx1250 async global->LDS copy (ASYNCcnt path), via inline asm: the clang
// builtin exists but types its pointer params in the CUDA __device__/__shared__
// language address spaces, which HIP source cannot spell. Inline asm bypasses
// that. VDST holds the workgroup-relative LDS byte address (HW adds LDS_BASE);
// the low 32 bits of a generic pointer to a __shared__ object are exactly that.
__device__ __forceinline__ void async_copy_b64(const void* gaddr, void* lds_ptr) {
  uint32_t loff = (uint32_t)(uintptr_t)lds_ptr;
  asm volatile("global_load_async_to_lds_b64 %0, %1, off"
               :
               : "v"(loff), "v"(gaddr)
               : "memory");
}

enum { MODE_DEG = 0, MODE_SMOOTH = 1 };

template <int MODE>
__device__ __forceinline__ void edge_op(int a, int b,
                                        const float* __restrict__ x,
                                        float* __restrict__ acc) {
  if (MODE == MODE_DEG) {
    atomicAdd(&acc[a], 1.0f);
    atomicAdd(&acc[b], 1.0f);
  } else {
    // gather both endpoints once, scatter to both (undirected edge)
    float bx = x[3 * b + 0], by = x[3 * b + 1], bz = x[3 * b + 2];
    float ax = x[3 * a + 0], ay = x[3 * a + 1], az = x[3 * a + 2];
    atomicAdd(&acc[3 * a + 0], bx);
    atomicAdd(&acc[3 * a + 1], by);
    atomicAdd(&acc[3 * a + 2], bz);
    atomicAdd(&acc[3 * b + 0], ax);
    atomicAdd(&acc[3 * b + 1], ay);
    atomicAdd(&acc[3 * b + 2], az);
  }
}

// Edge-stream kernel. Each wave (32 lanes) owns 32-edge chunks, grid-strided.
// Double-buffered global_load_async_to_lds_b64 of the next chunk overlaps with
// the random gathers + float atomics of the current chunk. Each wave touches
// only its own LDS slice, so no cross-wave barrier is needed — only the
// per-wave ASYNCcnt wait.
template <int MODE>
__global__ __launch_bounds__(256) void k_edge(const int* __restrict__ edges, int E,
                                              const float* __restrict__ x,
                                              float* __restrict__ acc) {
  __shared__ int2 se[8][2][32];  // 8 waves/block, double-buffered, 4 KB
  const int lane = threadIdx.x & 31;
  const int wid  = threadIdx.x >> 5;
  const int gw   = (int)blockIdx.x * 8 + wid;   // global wave id
  const int nw   = (int)gridDim.x * 8;          // total waves
  const int nch  = (E + 31) >> 5;               // 32-edge chunks
  if (gw < nch) {
    int gi = gw * 32 + lane;
    if (gi > E - 1) gi = E - 1;                 // clamp: redundant but valid
    async_copy_b64(edges + 2 * gi, &se[wid][0][lane]);
    int buf = 0;
    for (int c = gw; c < nch; c += nw, buf ^= 1) {
      int nxt = c + nw;
      int pc  = (nxt < nch) ? nxt : c;          // always issue: keeps ASYNCcnt
      int gi2 = pc * 32 + lane;                 // pattern uniform so wait(1)
      if (gi2 > E - 1) gi2 = E - 1;             // is always correct
      async_copy_b64(edges + 2 * gi2, &se[wid][buf ^ 1][lane]);
      WAIT_ASYNC(1);                            // current chunk resident in LDS
      int i = c * 32 + lane;
      if (i < E) {
        int2 e = se[wid][buf][lane];
        edge_op<MODE>(e.x, e.y, x, acc);
      }
    }
  }
}

// Zero deg + accumulator, convert faces to output dtype (one fused kernel).
__global__ __launch_bounds__(256) void k_setup(float* __restrict__ deg,
                                               float* __restrict__ acc, int V,
                                               const int* __restrict__ faces,
                                               float* __restrict__ outf, int F3) {
  int i = blockIdx.x * blockDim.x + threadIdx.x;
  if (i < V) deg[i] = 0.0f;
  if (i < 3 * V) acc[i] = 0.0f;
  if (i < F3) outf[i] = (float)faces[i];
}

__global__ __launch_bounds__(256) void k_invdeg(float* __restrict__ deg, int V) {
  int i = blockIdx.x * blockDim.x + threadIdx.x;
  if (i < V) deg[i] = 1.0f / fmaxf(deg[i], 1.0f);
}

// x_out = acc * inv_deg (lambda == 1.0 -> pure neighbor mean);
// optionally re-zero acc for the next smoothing pass (fused to save a launch).
__global__ __launch_bounds__(256) void k_scale(float* __restrict__ acc,
                                               const float* __restrict__ invd,
                                               float* __restrict__ xout, int V,
                                               int clear) {
  int i = blockIdx.x * blockDim.x + threadIdx.x;
  if (i < V) {
    float w = invd[i];
#pragma unroll
    for (int c = 0; c < 3; ++c) {
      xout[3 * i + c] = acc[3 * i + c] * w;
      if (clear) acc[3 * i + c] = 0.0f;
    }
  }
}

extern "C" void kernel_launch(void* const* d_in, const int* in_sizes, int n_in,
                              void* d_out, int out_size, void* d_ws, size_t ws_size,
                              hipStream_t stream) {
  const float* v     = (const float*)d_in[0];  // [1,V,3] f32
  const int*   edges = (const int*)d_in[1];    // [E,2] i32
  const int*   faces = (const int*)d_in[2];    // [1,F,3] i32

  const int V  = in_sizes[0] / 3;
  const int E  = in_sizes[1] / 2;
  const int F3 = in_sizes[2];

  float* deg = (float*)d_ws;      // V floats: degree, then 1/max(deg,1)
  float* acc = deg + V;           // 3V floats: neighbor-sum accumulator
  float* out_v = (float*)d_out;           // first 3V: smoothed verts
  float* out_f = out_v + (size_t)3 * V;   // next F3: faces (as output dtype)

  const int B = 256;
  auto g = [](int n) { return (n + 255) / 256; };

  // grid for edge-stream kernels: 8 waves/block over 32-edge chunks
  int nch = (E + 31) / 32;
  int eblocks = (nch + 7) / 8;
  if (eblocks > 4096) eblocks = 4096;
  if (eblocks < 1) eblocks = 1;

  int nSetup = (3 * V > F3) ? 3 * V : F3;
  k_setup<<<g(nSetup), B, 0, stream>>>(deg, acc, V, faces, out_f, F3);

  // degree via float atomics (acc arg = deg buffer)
  k_edge<MODE_DEG><<<eblocks, B, 0, stream>>>(edges, E, nullptr, deg);
  k_invdeg<<<g(V), B, 0, stream>>>(deg, V);

  // pass 1: acc = A * v ; out_v = acc * Dinv ; acc = 0
  k_edge<MODE_SMOOTH><<<eblocks, B, 0, stream>>>(edges, E, v, acc);
  k_scale<<<g(V), B, 0, stream>>>(acc, deg, out_v, V, 1);

  // pass 2: acc = A * out_v ; out_v = acc * Dinv
  k_edge<MODE_SMOOTH><<<eblocks, B, 0, stream>>>(edges, E, out_v, acc);
  k_scale<<<g(V), B, 0, stream>>>(acc, deg, out_v, V, 0);
}